// Spline2D_80118319940078
// MI455X (gfx1250) — compile-verified
//
#include <hip/hip_runtime.h>

// Spline2D on MI455X (gfx1250):
//   out[b,R,C] = sum_{i,j} bx[b,i]*by[b,j]*coeffs[R-px-96+i+2, C-py-96+j+2]
//   inside a 65x65 patch, 0 elsewhere.
// Bandwidth-bound: 256 MB output writes (~11us floor @ 23.3 TB/s).
// Kernel 1: non-temporal 256 MB zero fill (output > 192 MB L2).
// Kernel 2: template blend as a real matmul Templates(1024x4225) = W(1024x16) @ S(16x4225)
//           using V_WMMA_F32_16X16X4_F32 (K=16 -> 4 chained WMMAs/tile),
//           scatter-storing accumulators straight into the patch.

typedef __attribute__((ext_vector_type(2))) float v2f;
typedef __attribute__((ext_vector_type(4))) float v4f;
typedef __attribute__((ext_vector_type(8))) float v8f;

#define CW   72            // coeffs is 72x72
#define NT   65            // template edge (TH+1)
#define NRC  (NT * NT)     // 4225 template elements
#define NB   1024
#define OHW  (256 * 256)

__device__ __forceinline__ void bspline3(float p, float w[4]) {
    float p2 = p * p, p3 = p2 * p;
    w[0] = p3 * (1.0f / 6.0f);
    w[1] = -0.5f * p3 + 0.5f * p2 + 0.5f * p + (1.0f / 6.0f);
    w[2] = 0.5f * p3 - p2 + (2.0f / 3.0f);
    float q = 1.0f - p;
    w[3] = q * q * q * (1.0f / 6.0f);
}

// ---- Kernel 1: stream zeros with non-temporal b128 stores -------------------
__global__ __launch_bounds__(256) void fill_zero_nt(float* __restrict__ out, int n4) {
    int i = blockIdx.x * blockDim.x + threadIdx.x;
    if (i < n4) {
        v4f z = {0.0f, 0.0f, 0.0f, 0.0f};
        __builtin_nontemporal_store(z, ((v4f*)out) + i);
    }
}

// ---- Kernel 2: WMMA template blend + patch scatter --------------------------
// grid = (ceil(4225/16)=265, 1024/16=64), block = 32 (one wave per 16x16 tile)
__global__ __launch_bounds__(32) void spline_patch_wmma(
    const float* __restrict__ coeffs,
    const float* __restrict__ xs,
    const float* __restrict__ ys,
    float* __restrict__ out)
{
    const int lane = threadIdx.x;   // 0..31
    const int m    = lane & 15;     // A-row (batch-in-tile) and B-col (rc-in-tile)
    const int hi   = lane >> 4;     // lane half selects K pair {2hi, 2hi+1}

    // ---- A side: this lane's batch spline weights (W row) ----
    const int bA = blockIdx.y * 16 + m;
    float xv = xs[bA], yv = ys[bA];
    float sx = xv - floorf(xv);     // remainder(x, 1) == x - floor(x)
    float sy = yv - floorf(yv);
    float bx[4], by[4];
    bspline3(sx, bx);
    bspline3(sy, by);
    const float byA = hi ? by[2] : by[0];
    const float byB = hi ? by[3] : by[1];

    // ---- B side: this lane's template element rc ----
    const int rc_raw = blockIdx.x * 16 + m;
    const bool valid = rc_raw < NRC;
    const int rc = valid ? rc_raw : (NRC - 1);   // clamp so coeff loads stay in-bounds
    const int r = rc / NT;
    const int c = rc - r * NT;

    // k = i*4 + j ; i = s, j = 2hi(+1)  =>  B pair is two adjacent coeffs in row r+s+2
    const float* base = coeffs + (r + 2) * CW + (c + 2 * hi + 2);

    v8f acc = {};
#pragma unroll
    for (int s = 0; s < 4; ++s) {
        v2f a;
        a.x = bx[s] * byA;            // W[m, 4s+2hi]
        a.y = bx[s] * byB;            // W[m, 4s+2hi+1]
        v2f bv;
        bv.x = base[s * CW + 0];      // S[4s+2hi  , rc]
        bv.y = base[s * CW + 1];      // S[4s+2hi+1, rc]
        // D = A(16x4) * B(4x16) + C   (EXEC all-ones here: no divergence yet)
        acc = __builtin_amdgcn_wmma_f32_16x16x4_f32(
            /*neg_a=*/false, a, /*neg_b=*/false, bv,
            /*c_mod=*/(short)0, acc, /*reuse_a=*/false, /*reuse_b=*/false);
    }

    // ---- Scatter: lane holds D for batches (v + 8*hi) at column rc ----
    if (valid) {
#pragma unroll
        for (int v = 0; v < 8; ++v) {
            int bb = blockIdx.y * 16 + hi * 8 + v;   // D-layout: M = v + 8*(lane/16)
            float xb = xs[bb], yb = ys[bb];
            int px = (int)floorf(xb);
            int py = (int)floorf(yb);
            int R = r + px + 96;
            int C = c + py + 96;
            if (R >= 0 && R < 256 && C >= 0 && C < 256) {
                out[(size_t)bb * OHW + (size_t)R * 256 + C] = acc[v];
            }
        }
    }
}

extern "C" void kernel_launch(void* const* d_in, const int* in_sizes, int n_in,
                              void* d_out, int out_size, void* d_ws, size_t ws_size,
                              hipStream_t stream) {
    const float* coeffs = (const float*)d_in[0];   // (72,72) f32
    const float* x      = (const float*)d_in[1];   // (1024,1,1,1) f32
    const float* y      = (const float*)d_in[2];   // (1024,1,1,1) f32
    float* out          = (float*)d_out;           // 1024*1*256*256 f32

    const int n4 = out_size / 4;                   // float4 count (out_size % 4 == 0)
    fill_zero_nt<<<(n4 + 255) / 256, 256, 0, stream>>>(out, n4);

    dim3 grid((NRC + 15) / 16, NB / 16);           // (265, 64) tiles
    spline_patch_wmma<<<grid, 32, 0, stream>>>(coeffs, x, y, out);
}